// Aggregator_40114994545368
// MI455X (gfx1250) — compile-verified
//
#include <hip/hip_runtime.h>

typedef __attribute__((ext_vector_type(2))) float v2f;
typedef __attribute__((ext_vector_type(8))) float v8f;

#define DIM 128
#define BETA 0.4054651081081644f   /* ln(1.5) */
#define ALPHA_ 0.1f
#define SLOPE 0.01f

// Swizzled-M layout: element (r,c) of branch b at (r>>1)*MPAIR + b*256 + 2c + (r&1)
#define MPAIR 544                  // floats per K-pair block (padded for bank spread)
#define MTOT  (64 * MPAIR)         // 34816 floats = 136KB
// X-tile layout: row i of x_sum at sX + i*XROW, x_bi at +128; pad to 260 floats
#define XROW  260
#define XTILE (16 * XROW)          // 4160 floats per wave

// ---------------------------------------------------------------------------
// Zero the SpMM accumulator (must run every call; harness doesn't re-poison).
// ---------------------------------------------------------------------------
__global__ void zero_side(float4* __restrict__ p, int n4) {
    int i = blockIdx.x * blockDim.x + threadIdx.x;
    if (i < n4) p[i] = make_float4(0.f, 0.f, 0.f, 0.f);
}

// ---------------------------------------------------------------------------
// M1 = ((1-beta) + beta*W) @ w1 ; M2 = ((1-beta) + beta*W) @ w2
// ---------------------------------------------------------------------------
__global__ void prep(const float* __restrict__ W, const float* __restrict__ w1,
                     const float* __restrict__ w2, float* __restrict__ M1,
                     float* __restrict__ M2) {
    __shared__ float im[DIM];
    const int i = blockIdx.x;
    const int t = threadIdx.x;
    if (t < DIM) im[t] = (1.0f - BETA) + BETA * W[i * DIM + t];
    __syncthreads();
    const float* w = (t < DIM) ? w1 : w2;
    const int j = t & (DIM - 1);
    float acc = 0.f;
#pragma unroll 8
    for (int k = 0; k < DIM; ++k) acc = fmaf(im[k], w[k * DIM + j], acc);
    ((t < DIM) ? M1 : M2)[i * DIM + j] = acc;
}

// ---------------------------------------------------------------------------
// SpMM scatter: one wave32 per edge; coalesced 512B gather + 4 f32 atomics/lane.
// ---------------------------------------------------------------------------
__global__ void spmm(const float* __restrict__ ego, const float* __restrict__ vals,
                     const int* __restrict__ row, const int* __restrict__ col,
                     float* __restrict__ side, int E) {
    const int gid = blockIdx.x * blockDim.x + threadIdx.x;
    const int e = gid >> 5;
    if (e >= E) return;
    const int lane = gid & 31;
    const int c = col[e];
    const int r = row[e];
    const float v = vals[e];
    const float4 x = ((const float4*)(ego + (size_t)c * DIM))[lane];
    float* dst = side + (size_t)r * DIM + lane * 4;
    atomicAdd(dst + 0, v * x.x);
    atomicAdd(dst + 1, v * x.y);
    atomicAdd(dst + 2, v * x.z);
    atomicAdd(dst + 3, v * x.w);
}

// ---------------------------------------------------------------------------
// Fused: out = leaky(x_sum@M1 + b1) + leaky(x_bi@M2 + b2)
// 8 waves/block, one 16-row tile per wave. Two passes of 4 N-tiles so the
// accumulator working set is 64 VGPRs (no spills). All WMMA fragments are
// single contiguous ds_load_b64's thanks to the swizzled LDS layouts.
// ---------------------------------------------------------------------------
__global__ void fused(const float* __restrict__ ego, const float* __restrict__ h0,
                      const float* __restrict__ side, const float* __restrict__ M1,
                      const float* __restrict__ M2, const float* __restrict__ bias1,
                      const float* __restrict__ bias2, float* __restrict__ out,
                      int N, int ntiles) {
    extern __shared__ float smem[];
    float* sMsw = smem;                      // swizzled M1|M2, MTOT floats
    const int tid  = threadIdx.x;
    const int wave = tid >> 5;
    const int lane = tid & 31;
    float* sX = smem + MTOT + wave * XTILE;  // this wave's x tiles

    // Stage M1/M2 into swizzled LDS: fragment rows (2k,2k+1) become an
    // adjacent float pair -> B fragment = one aligned ds_load_b64.
    {
        const float4* g1 = (const float4*)M1;
        const float4* g2 = (const float4*)M2;
        for (int i = tid; i < DIM * DIM / 4; i += 256) {
            const float4 a = g1[i];
            const float4 b = g2[i];
            const int r = i >> 5;            // row, 32 float4 per row
            const int c0 = (i & 31) * 4;
            float* q = sMsw + (r >> 1) * MPAIR + c0 * 2 + (r & 1);
            q[0] = a.x; q[2] = a.y; q[4] = a.z; q[6] = a.w;
            q[256] = b.x; q[258] = b.y; q[260] = b.z; q[262] = b.w;
        }
    }
    __syncthreads();

    const int tile = blockIdx.x * 8 + wave;
    if (tile >= ntiles) return;              // wave-uniform; EXEC stays full
    const int r0 = tile * 16;
    const bool full = (r0 + 16 <= N);        // wave-uniform fast-path flag

    // Build x tiles. Lane covers cols [4*lane, 4*lane+4); rows interleaved:
    // x_sum at sX+i*XROW, x_bi at +128 (so af1/af2 merge to ds_load_2addr_b64).
    for (int i = 0; i < 16; ++i) {
        int r = r0 + i;
        int rc = (r < N) ? r : (N - 1);      // uniform clamp, no divergence
        const float4 e = ((const float4*)(ego  + (size_t)rc * DIM))[lane];
        const float4 s = ((const float4*)(side + (size_t)rc * DIM))[lane];
        const float4 h = ((const float4*)(h0   + (size_t)rc * DIM))[lane];
        float* ps = sX + i * XROW + 4 * lane;
        v2f t0 = {0.9f * (e.x + s.x) + ALPHA_ * h.x, 0.9f * (e.y + s.y) + ALPHA_ * h.y};
        v2f t1 = {0.9f * (e.z + s.z) + ALPHA_ * h.z, 0.9f * (e.w + s.w) + ALPHA_ * h.w};
        v2f u0 = {0.9f * (e.x * s.x) + ALPHA_ * h.x, 0.9f * (e.y * s.y) + ALPHA_ * h.y};
        v2f u1 = {0.9f * (e.z * s.z) + ALPHA_ * h.z, 0.9f * (e.w * s.w) + ALPHA_ * h.w};
        *(v2f*)(ps)       = t0;
        *(v2f*)(ps + 2)   = t1;
        *(v2f*)(ps + 128) = u0;
        *(v2f*)(ps + 130) = u1;
    }
    // Wave-local LDS RAW: LDS ops in-order per wave; compiler adds s_wait_dscnt.

    // WMMA f32 16x16x4 fragment addressing (ISA 7.12.2):
    //   A: lane L holds row M=L[3:0], K = {k+2*L[4], k+1+2*L[4]}
    //   B: lane L holds col N=n0+L[3:0], same K pair
    //   C/D: VGPR v -> row v+8*L[4], col n0+L[3:0]
    const int half = lane >> 4;
    const int l15  = lane & 15;

#pragma unroll 1
    for (int pass = 0; pass < 2; ++pass) {
        v8f acc1[4], acc2[4];
#pragma unroll
        for (int nt = 0; nt < 4; ++nt) { acc1[nt] = (v8f){}; acc2[nt] = (v8f){}; }

#pragma unroll 1
        for (int k = 0; k < DIM; k += 4) {
            const float* ap = sX + l15 * XROW + k + 2 * half;
            const v2f af1 = *(const v2f*)(ap);        // x_sum frag  \ one
            const v2f af2 = *(const v2f*)(ap + 128);  // x_bi frag   / 2addr_b64
            const float* bp = sMsw + (k / 2 + half) * MPAIR + pass * 128 + l15 * 2;
#pragma unroll
            for (int nt = 0; nt < 4; ++nt) {
                const v2f bf1 = *(const v2f*)(bp + nt * 32);        // M1 frag
                const v2f bf2 = *(const v2f*)(bp + nt * 32 + 256);  // M2 frag
                acc1[nt] = __builtin_amdgcn_wmma_f32_16x16x4_f32(
                    false, af1, false, bf1, (short)0, acc1[nt], false, false);
                acc2[nt] = __builtin_amdgcn_wmma_f32_16x16x4_f32(
                    false, af2, false, bf2, (short)0, acc2[nt], false, false);
            }
        }

        // Epilogue: bias + LeakyReLU per branch, sum, store.
        // Wave-uniform fast path: full tiles store all 64 elements unguarded
        // (avoids the per-store saveexec ladder); only a tail tile predicates.
#pragma unroll
        for (int nt = 0; nt < 4; ++nt) {
            const int n0 = pass * 64 + nt * 16;
            const float bb1 = bias1[n0 + l15];
            const float bb2 = bias2[n0 + l15];
            float res[8];
#pragma unroll
            for (int v = 0; v < 8; ++v) {
                float s1 = acc1[nt][v] + bb1;
                s1 = (s1 >= 0.f) ? s1 : SLOPE * s1;
                float s2 = acc2[nt][v] + bb2;
                s2 = (s2 >= 0.f) ? s2 : SLOPE * s2;
                res[v] = s1 + s2;
            }
            float* op = out + (size_t)(r0 + 8 * half) * DIM + n0 + l15;
            if (full) {
#pragma unroll
                for (int v = 0; v < 8; ++v) op[(size_t)v * DIM] = res[v];
            } else {
#pragma unroll
                for (int v = 0; v < 8; ++v)
                    if (r0 + v + 8 * half < N) op[(size_t)v * DIM] = res[v];
            }
        }
    }
}

// ---------------------------------------------------------------------------
extern "C" void kernel_launch(void* const* d_in, const int* in_sizes, int n_in,
                              void* d_out, int out_size, void* d_ws, size_t ws_size,
                              hipStream_t stream) {
    const float* ego  = (const float*)d_in[0];
    const float* h0   = (const float*)d_in[1];
    const float* vals = (const float*)d_in[2];
    const float* W    = (const float*)d_in[3];
    const float* w1   = (const float*)d_in[4];
    const float* b1   = (const float*)d_in[5];
    const float* w2   = (const float*)d_in[6];
    const float* b2   = (const float*)d_in[7];
    const int*   row  = (const int*)d_in[8];
    const int*   col  = (const int*)d_in[9];

    const int N = in_sizes[0] / DIM;    // 100000
    const int E = in_sizes[8];          // 3200000

    // Workspace layout: side (N*128 f32) | M1 (128*128) | M2 (128*128)
    float* side = (float*)d_ws;
    float* M1   = side + (size_t)N * DIM;
    float* M2   = M1 + DIM * DIM;

    const int n4 = N * (DIM / 4);
    zero_side<<<(n4 + 255) / 256, 256, 0, stream>>>((float4*)side, n4);
    prep<<<DIM, 256, 0, stream>>>(W, w1, w2, M1, M2);
    const long long spmm_threads = (long long)E * 32;
    spmm<<<(unsigned)((spmm_threads + 255) / 256), 256, 0, stream>>>(
        ego, vals, row, col, side, E);
    const int ntiles = (N + 15) / 16;
    // LDS: swizzled M (136KB) + 8 waves * 16.25KB x-tiles = 266KB (<320KB/WGP).
    const size_t lds = (MTOT + 8 * XTILE) * sizeof(float);
    fused<<<(ntiles + 7) / 8, 256, lds, stream>>>(
        ego, h0, side, M1, M2, b1, b2, (float*)d_out, N, ntiles);
}